// HaarWaveletTransform_18932215841283
// MI455X (gfx1250) — compile-verified
//
#include <hip/hip_runtime.h>

// Haar DWT forward, x:(8,64,512,512) f32 -> (low, high) each (8,64,256,256) f32.
// low  = 0.5*(a+b+c+d)
// high = lh+hl+hh = 0.5*(3d - a - b - c)
//
// Pure streaming problem: 768 MiB mandatory traffic -> ~34us floor @ 23.3 TB/s.
// Strategy: b128 coalesced accesses, gfx1250 async global->LDS staging
// (GLOBAL_LOAD_ASYNC_TO_LDS_B128 + s_wait_asynccnt), non-temporal b128 stores.

typedef float v4f __attribute__((ext_vector_type(4)));

#define W_IN    512u
#define PLANES  512u           // 8 * 64
#define OW      256u
#define OH      256u

__global__ __launch_bounds__(256)
void haar_dwt_18932215841283_kernel(const float* __restrict__ x,
                                    float* __restrict__ lo_out,
                                    float* __restrict__ hi_out) {
    const uint32_t tid = blockIdx.x * 256u + threadIdx.x;
    const uint32_t tx  = tid & 63u;           // 64 groups of 4 output cols
    const uint32_t row = (tid >> 6) & 255u;   // output row
    const uint32_t pl  = tid >> 14;           // plane (n*64 + c), 0..511

    // input row pair base: plane*512*512 + (2*row)*512 + tx*8
    const float* r0 = x + ((size_t)pl << 18) + ((size_t)row << 10) + ((size_t)tx << 3);
    const float* r1 = r0 + W_IN;

    v4f e0, e1, o0, o1; // e*: even row (a,b interleaved), o*: odd row (c,d interleaved)

#if defined(__AMDGCN__)
    // Stage both input rows through LDS with the CDNA5 async copy path.
    // 64 B per thread -> 16 KB per 256-thread block.
    __shared__ float lds[256 * 16];
    float* slot_p = &lds[threadIdx.x * 16u];
    // low 32 bits of a generic shared pointer == LDS byte offset (aperture in hi32)
    const uint32_t slot0 = (uint32_t)(uintptr_t)slot_p;        // row0: bytes [0,32)
    const uint32_t slot1 = slot0 + 32u;                        // row1: bytes [32,64)
    // NOTE: inst offset is added to BOTH the global and the LDS address,
    // which matches the contiguous 32B-per-row slot layout.
    asm volatile(
        "global_load_async_to_lds_b128 %0, %2, off\n\t"
        "global_load_async_to_lds_b128 %0, %2, off offset:16\n\t"
        "global_load_async_to_lds_b128 %1, %3, off\n\t"
        "global_load_async_to_lds_b128 %1, %3, off offset:16\n\t"
        "s_wait_asynccnt 0"
        :
        : "v"(slot0), "v"(slot1), "v"(r0), "v"(r1)
        : "memory");
    const v4f* s = (const v4f*)slot_p;   // ds_load_b128 x4, own slot only
    e0 = s[0]; e1 = s[1]; o0 = s[2]; o1 = s[3];
#else
    // host-side parse path / non-AMDGCN fallback
    e0 = __builtin_nontemporal_load((const v4f*)r0);
    e1 = __builtin_nontemporal_load((const v4f*)(r0 + 4));
    o0 = __builtin_nontemporal_load((const v4f*)r1);
    o1 = __builtin_nontemporal_load((const v4f*)(r1 + 4));
#endif

    // e0 = (a0,b0,a1,b1)  e1 = (a2,b2,a3,b3)
    // o0 = (c0,d0,c1,d1)  o1 = (c2,d2,c3,d3)
    v4f lo, hi;
    {
        float s0 = e0.x + e0.y + o0.x;  float d0 = o0.y;
        float s1 = e0.z + e0.w + o0.z;  float d1 = o0.w;
        float s2 = e1.x + e1.y + o1.x;  float d2 = o1.y;
        float s3 = e1.z + e1.w + o1.z;  float d3 = o1.w;
        lo.x = 0.5f * (s0 + d0);  hi.x = 1.5f * d0 - 0.5f * s0;
        lo.y = 0.5f * (s1 + d1);  hi.y = 1.5f * d1 - 0.5f * s1;
        lo.z = 0.5f * (s2 + d2);  hi.z = 1.5f * d2 - 0.5f * s2;
        lo.w = 0.5f * (s3 + d3);  hi.w = 1.5f * d3 - 0.5f * s3;
    }

    // output: plane*256*256 + row*256 + tx*4 ; write-once -> non-temporal b128
    const size_t obase = ((size_t)pl << 16) + ((size_t)row << 8) + ((size_t)tx << 2);
    __builtin_nontemporal_store(lo, (v4f*)(lo_out + obase));
    __builtin_nontemporal_store(hi, (v4f*)(hi_out + obase));
}

extern "C" void kernel_launch(void* const* d_in, const int* in_sizes, int n_in,
                              void* d_out, int out_size, void* d_ws, size_t ws_size,
                              hipStream_t stream) {
    (void)in_sizes; (void)n_in; (void)d_ws; (void)ws_size;
    const float* x  = (const float*)d_in[0];
    float*       lo = (float*)d_out;
    float*       hi = lo + (size_t)out_size / 2;   // low || high, concatenated flat

    // 512 planes * 256 rows * 64 col-groups = 8,388,608 threads
    const uint32_t total_threads = PLANES * OH * (OW / 4);
    dim3 block(256);
    dim3 grid(total_threads / 256);
    haar_dwt_18932215841283_kernel<<<grid, block, 0, stream>>>(x, lo, hi);
}